// NormConvTranspose2d_o_21328807592462
// MI455X (gfx1250) — compile-verified
//
#include <hip/hip_runtime.h>

// NormConvTranspose2d + InstanceNorm for MI455X (gfx1250, wave32).
//
// out[b,i,y,x] = sum_c convT(x[b,c], W[c,i])[y,x] / (convT(ones, W[c,i])[y,x] + 1e-10)
// then InstanceNorm2d over (y,x).  Bias cancels under InstanceNorm -> dropped.
//
// convT(stride=2,K=4,pad=1): out[y,x] gets taps ky = y+1-2h. Interior even y:
// ky in {1,3}; odd y: {0,2}; y==0: {1}; y==95: {2} (same in x). So the
// normalizer is constant per (c,i, 4x4 position-class), and interior outputs
// are a GEMM over K = Cin x (2x2 taps) with class-normalized weights ->
// mapped onto V_WMMA_F32_16X16X4_F32 (full fp32 precision; problem is ~0.6
// GFLOP / ~6 MB so it is bandwidth/latency bound at 23.3 TB/s regardless).

#define EPS_F 1e-10f
#define IN_EPS_F 1e-5f

typedef __attribute__((ext_vector_type(2))) float v2f;
typedef __attribute__((ext_vector_type(8))) float v8f;

static constexpr int CIN = 64, COUT = 64, H = 48, WD = 48, HO = 96, WO = 96;
static constexpr int CH = H * WD;          // 2304 floats per input channel
static constexpr int RING = 380;           // 96*96 - 94*94 border pixels

// ---------------------------------------------------------------------------
// Kernel 1: per (c,i) compute den for all 16 position classes, and the
// normalized weights for the 4 interior parity classes, pre-swizzled into the
// WMMA B-matrix layout: within each (class,tap) 64x64 block,
//   idx = (chunk*2 + khalf)*128 + i*2 + j   with  c = chunk*4 + khalf*2 + j
// so lane l loads float2 at (chunk*2 + (l>>4))*128 + (n0 + (l&15))*2.
// ---------------------------------------------------------------------------
__global__ void nct_precompute(const float* __restrict__ W,
                               float* __restrict__ Wn,
                               float* __restrict__ den_all) {
  int t = blockIdx.x * blockDim.x + threadIdx.x;
  if (t >= CIN * COUT) return;
  int c = t >> 6, i = t & 63;
  const float* w = W + ((c * 64 + i) << 4);   // [ky][kx], 16 floats
  float wv[16];
#pragma unroll
  for (int q = 0; q < 16; ++q) wv[q] = w[q];

  // tap sets per class: 0:{1}  1:{1,3}  2:{0,2}  3:{2}
  const int set0[4] = {1, 1, 0, 2};
  const int set1[4] = {-1, 3, 2, -1};

  float den[4][4];
  for (int yc = 0; yc < 4; ++yc)
    for (int xc = 0; xc < 4; ++xc) {
      float s = 0.f;
      int kys[2] = {set0[yc], set1[yc]};
      int kxs[2] = {set0[xc], set1[xc]};
      for (int a = 0; a < 2; ++a)
        if (kys[a] >= 0)
          for (int bb = 0; bb < 2; ++bb)
            if (kxs[bb] >= 0) s += wv[kys[a] * 4 + kxs[bb]];
      den[yc][xc] = s + EPS_F;
      den_all[(yc * 4 + xc) * 4096 + c * 64 + i] = den[yc][xc];
    }

  // interior parity classes: py=0 -> ycls 1 (ky: ty0->1, ty1->3)
  //                          py=1 -> ycls 2 (ky: ty0->0, ty1->2)
  int chunk = c >> 2, kh = (c >> 1) & 1, j = c & 1;
  int widx = (chunk * 2 + kh) * 128 + i * 2 + j;
  for (int py = 0; py < 2; ++py)
    for (int px = 0; px < 2; ++px) {
      float d = den[py ? 2 : 1][px ? 2 : 1];
      for (int ty = 0; ty < 2; ++ty)
        for (int tx = 0; tx < 2; ++tx) {
          int ky = py ? (ty ? 2 : 0) : (ty ? 3 : 1);
          int kx = px ? (tx ? 2 : 0) : (tx ? 3 : 1);
          int blk = (py * 2 + px) * 4 + ty * 2 + tx;
          Wn[blk * 4096 + widx] = wv[ky * 4 + kx] / d;
        }
    }
}

// ---------------------------------------------------------------------------
// Kernel 2: WMMA main kernel. One wave per 16-pixel x 16-channel tile.
// M = 16 same-parity x positions of one output row, N = 16 out channels,
// K = 64 Cin x 4 taps = 256  ->  64x V_WMMA_F32_16X16X4_F32.
// ---------------------------------------------------------------------------
__global__ void __launch_bounds__(32)
nct_main_wmma(const float* __restrict__ x, const float* __restrict__ Wn,
              float* __restrict__ out) {
  int bid = blockIdx.x;
  int nt = bid & 3;  bid >>= 2;            // out-channel tile
  int xtp = bid % 6; bid /= 6;             // (px, xt)
  int y = bid % 96;  int b = bid / 96;
  int px = xtp / 3, xt = xtp % 3;
  int py = y & 1;

  int l = threadIdx.x;
  int m = l & 15, kh = l >> 4;

  int xm = px + 2 * (16 * xt + m);         // this lane's output x (A-role)
  int wA = (xm + px) >> 1;                 // tap tx=0 input col; tx=1 is wA-1
  int h0 = (y + py) >> 1;                  // tap ty=0 input row; ty=1 is h0-1
  int n0 = nt * 16;

  const float* xb = x + b * CIN * CH;
  const float* wb = Wn + ((py * 2 + px) * 4) * 4096;

  int hs[2]  = {h0, h0 - 1};
  int wss[2] = {wA, wA - 1};
  bool hok[2] = {(unsigned)hs[0] < 48u, (unsigned)hs[1] < 48u};
  bool wok[2] = {(unsigned)wss[0] < 48u, (unsigned)wss[1] < 48u};

  v8f acc = {};
  for (int chunk = 0; chunk < 16; ++chunk) {
    int cbase = chunk * 4 + kh * 2;        // A lane holds cin = cbase, cbase+1
    const float* xc = xb + cbase * CH;
#pragma unroll
    for (int ty = 0; ty < 2; ++ty) {
#pragma unroll
      for (int tx = 0; tx < 2; ++tx) {
        bool ok = hok[ty] && wok[tx];
        int off = ok ? (hs[ty] * WD + wss[tx]) : 0;   // clamp: always valid addr
        float v0 = xc[off];
        float v1 = xc[CH + off];
        v2f a; a.x = ok ? v0 : 0.f; a.y = ok ? v1 : 0.f;
        const float* bp = wb + (ty * 2 + tx) * 4096 +
                          (chunk * 2 + kh) * 128 + (n0 + m) * 2;
        v2f bm; bm.x = bp[0]; bm.y = bp[1];
        // 8 args: (neg_a, A, neg_b, B, c_mod, C, reuse_a, reuse_b)
        acc = __builtin_amdgcn_wmma_f32_16x16x4_f32(
            false, a, false, bm, (short)0, acc, false, false);
      }
    }
  }

  // D layout: lanes 0-15 hold M=v, lanes 16-31 hold M=v+8; N = lane&15.
  float* ob = out + ((size_t)(b * 64 + n0 + m) * 96 + y) * 96;
#pragma unroll
  for (int v = 0; v < 8; ++v) {
    int Mv = v + 8 * kh;
    int xo = px + 2 * (16 * xt + Mv);
    ob[xo] = acc[v];
  }
}

// ---------------------------------------------------------------------------
// Kernel 3: recompute the 1-pixel border ring with the exact per-class den
// (the main kernel used interior-class normalization there). 380 px/image.
// ---------------------------------------------------------------------------
__global__ void nct_fixup(const float* __restrict__ x, const float* __restrict__ W,
                          const float* __restrict__ den_all,
                          float* __restrict__ out) {
  int t = blockIdx.x * blockDim.x + threadIdx.x;
  if (t >= 2 * COUT * RING) return;
  int r = t % RING; int bi = t / RING;
  int i = bi & 63, b = bi >> 6;

  int y, xo;
  if (r < 96)       { y = 0;  xo = r; }
  else if (r < 192) { y = 95; xo = r - 96; }
  else { int rr = r - 192; int side = rr / 94; y = 1 + (rr % 94); xo = side ? 95 : 0; }

  int ycls = (y == 0) ? 0 : (y == 95) ? 3 : ((y & 1) ? 2 : 1);
  int xcls = (xo == 0) ? 0 : (xo == 95) ? 3 : ((xo & 1) ? 2 : 1);
  const int set0[4] = {1, 1, 0, 2};
  const int set1[4] = {-1, 3, 2, -1};
  int kys[2] = {set0[ycls], set1[ycls]};
  int kxs[2] = {set0[xcls], set1[xcls]};

  const float* dd = den_all + (ycls * 4 + xcls) * 4096 + i;
  const float* xb = x + b * CIN * CH;
  float acc = 0.f;
  for (int c = 0; c < CIN; ++c) {
    const float* wv = W + ((c * 64 + i) << 4);
    const float* xc = xb + c * CH;
    float num = 0.f;
    for (int a = 0; a < 2; ++a) {
      int ky = kys[a]; if (ky < 0) continue;
      int h = (y + 1 - ky) >> 1;                 // in-range by class construction
      for (int bb = 0; bb < 2; ++bb) {
        int kx = kxs[bb]; if (kx < 0) continue;
        int w = (xo + 1 - kx) >> 1;
        num += xc[h * WD + w] * wv[ky * 4 + kx];
      }
    }
    acc += num / dd[c * 64];
  }
  out[((size_t)(b * 64 + i) * 96 + y) * 96 + xo] = acc;
}

// ---------------------------------------------------------------------------
// Kernel 4: in-place InstanceNorm2d per (b,c). 9216 spatial elements.
// Bias omitted: constant-per-channel shift cancels under mean subtraction.
// ---------------------------------------------------------------------------
__global__ void nct_instnorm(float* __restrict__ out) {
  __shared__ float ss[256], ss2[256];
  float* p = out + (size_t)blockIdx.x * (HO * WO);
  float s = 0.f, s2 = 0.f;
  for (int k = threadIdx.x; k < HO * WO; k += 256) {
    float v = p[k]; s += v; s2 += v * v;
  }
  ss[threadIdx.x] = s; ss2[threadIdx.x] = s2;
  __syncthreads();
  for (int st = 128; st > 0; st >>= 1) {
    if (threadIdx.x < st) {
      ss[threadIdx.x] += ss[threadIdx.x + st];
      ss2[threadIdx.x] += ss2[threadIdx.x + st];
    }
    __syncthreads();
  }
  float mean = ss[0] * (1.f / (HO * WO));
  float var = ss2[0] * (1.f / (HO * WO)) - mean * mean;
  float inv = rsqrtf(var + IN_EPS_F);
  for (int k = threadIdx.x; k < HO * WO; k += 256)
    p[k] = (p[k] - mean) * inv;
}

extern "C" void kernel_launch(void* const* d_in, const int* in_sizes, int n_in,
                              void* d_out, int out_size, void* d_ws, size_t ws_size,
                              hipStream_t stream) {
  const float* x = (const float*)d_in[0];      // [2,64,48,48]
  const float* W = (const float*)d_in[1];      // [64,64,4,4]
  // d_in[2] = bias: cancels under InstanceNorm (affine=False) -> unused.
  float* out = (float*)d_out;                  // [2,64,96,96]

  float* Wn      = (float*)d_ws;               // 65536 floats (256 KB)
  float* den_all = Wn + 4 * 4 * 64 * 64;       // 65536 floats (256 KB)

  nct_precompute<<<16, 256, 0, stream>>>(W, Wn, den_all);
  nct_main_wmma<<<2 * 96 * 6 * 4, 32, 0, stream>>>(x, Wn, out);
  nct_fixup<<<(2 * COUT * RING + 255) / 256, 256, 0, stream>>>(x, W, den_all, out);
  nct_instnorm<<<2 * COUT, 256, 0, stream>>>(out);
}